// HyperConnections_66245575573943
// MI455X (gfx1250) — compile-verified
//
#include <hip/hip_runtime.h>
#include <hip/hip_bf16.h>
#include <math.h>
#include <stdint.h>

typedef float v2f __attribute__((ext_vector_type(2)));
typedef float v8f __attribute__((ext_vector_type(8)));

#define S_      4
#define D_      1024
#define SD      4096            // S*D
#define NT      8192            // B*T tokens
#define NCH     24              // 16 res + 4 pre + 4 post channels
#define CH      256             // K chunk staged in LDS
#define NCHUNK  (SD / CH)       // 16
#define PAD     4
#define LDW     (CH + PAD)      // 260 -> strided A reads hit distinct banks
#define KPAIRS  (SD / 2)        // 2048
#define WPACK_FLOATS (KPAIRS * 32 * 2)   // 131072 floats = 512 KB

// ---------------------------------------------------------------------------
// CDNA5 async global->LDS copy (ASYNCcnt-tracked), 16 bytes per lane.
// VDST = VGPR holding LDS byte address, VADDR = VGPR pair with global address.
// ---------------------------------------------------------------------------
__device__ __forceinline__ void async_cp16(const float* g, float* l) {
  asm volatile("global_load_async_to_lds_b128 %0, %1, off"
               :
               : "v"((unsigned)(size_t)l), "v"(g)
               : "memory");
}
__device__ __forceinline__ void wait_asynccnt_8() {
  asm volatile("s_wait_asynccnt 0x8" ::: "memory");
}
__device__ __forceinline__ void wait_asynccnt_0() {
  asm volatile("s_wait_asynccnt 0x0" ::: "memory");
}

// ---------------------------------------------------------------------------
// Kernel 0: fold (gamma+1) into combined weights, pair-packed for WMMA B.
// wpack[p][n][q] = W_n[2p+q] * (gamma[2p+q]+1),  n<16: w_res row n,
// 16..19: w_pre, 20..23: w_post, 24..31: zero padding.
// ---------------------------------------------------------------------------
__global__ __launch_bounds__(256) void k_fold(const float* __restrict__ gamma,
                                              const float* __restrict__ w_res,
                                              const float* __restrict__ w_pre,
                                              const float* __restrict__ w_post,
                                              float* __restrict__ wpack) {
  int idx = blockIdx.x * 256 + threadIdx.x;   // p*32 + n
  if (idx >= KPAIRS * 32) return;
  int p = idx >> 5;
  int n = idx & 31;
  float out2[2];
#pragma unroll
  for (int q = 0; q < 2; ++q) {
    int k = p * 2 + q;
    float g = gamma[k] + 1.0f;
    float w = 0.0f;
    if (n < 16)       w = w_res[n * SD + k];
    else if (n < 20)  w = w_pre[(n - 16) * SD + k];
    else if (n < 24)  w = w_post[(n - 20) * SD + k];
    out2[q] = w * g;
  }
  *(float2*)&wpack[(size_t)idx * 2] = make_float2(out2[0], out2[1]);
}

// ---------------------------------------------------------------------------
// Kernel 1: per 16-token tile, sum-of-squares + 24 dot products via
// V_WMMA_F32_16X16X4_F32 with double-buffered async global->LDS staging,
// then Sinkhorn/softmax/sigmoid -> hbuf[token][24].
// ---------------------------------------------------------------------------
__device__ __forceinline__ float lse4(float a, float b, float c, float d) {
  float m = fmaxf(fmaxf(a, b), fmaxf(c, d));
  return m + logf(expf(a - m) + expf(b - m) + expf(c - m) + expf(d - m));
}

__global__ __launch_bounds__(128) void k_stats(const float* __restrict__ resid,
                                               const float* __restrict__ wpack,
                                               const float* __restrict__ beta_res,
                                               const float* __restrict__ beta_pre,
                                               const float* __restrict__ beta_post,
                                               const float* __restrict__ alpha_res,
                                               const float* __restrict__ alpha_pre,
                                               const float* __restrict__ alpha_post,
                                               float* __restrict__ hbuf) {
  __shared__ float tileA[2][16 * LDW];    // double-buffered 16 tokens x 256 K
  __shared__ float cred[2 * 256];         // reduced C: [tile(2)][M(16)][N(16)]
  __shared__ float ssq[16];               // per-token sum of squares

  const int tid  = threadIdx.x;
  const int lane = tid & 31;
  const int wave = tid >> 5;              // 0..3
  const size_t tok0 = (size_t)blockIdx.x * 16;

  for (int i = tid; i < 512; i += 128) cred[i] = 0.0f;
  if (tid < 16) ssq[tid] = 0.0f;

  // staging role: 8 threads per token, 32 floats (2 x b128 ... 8 x b128) each
  const int tk  = tid >> 3;               // token within tile
  const int sub = tid & 7;
  const float* gsrc = resid + (tok0 + tk) * SD + sub * 32;
  float* lst[2] = { &tileA[0][tk * LDW + sub * 32],
                    &tileA[1][tk * LDW + sub * 32] };

  // WMMA lane roles (16x4 f32 A layout / 4x16 f32 B layout)
  const int aTok   = lane & 15;           // M = token row
  const int kShift = (lane >> 4) * 2;     // lanes 0-15 -> K0,K1 ; 16-31 -> K2,K3
  const int bN     = lane & 15;           // N column within tile

  // prologue: async-stage chunk 0 into buffer 0 (8 x 16B per thread)
#pragma unroll
  for (int i = 0; i < 8; ++i) async_cp16(gsrc + i * 4, lst[0] + i * 4);

  float my_ssq = 0.0f;
  v8f c0 = {};                            // channels 0..15
  v8f c1 = {};                            // channels 16..23 (N tile 16..31)

  for (int cnk = 0; cnk < NCHUNK; ++cnk) {
    // issue next chunk's async copies into the other buffer, then wait for
    // the current chunk's 8 copies (in-order completion) before the barrier
    if (cnk + 1 < NCHUNK) {
      const float* g = gsrc + (cnk + 1) * CH;
      float* l = lst[(cnk + 1) & 1];
#pragma unroll
      for (int i = 0; i < 8; ++i) async_cp16(g + i * 4, l + i * 4);
      wait_asynccnt_8();
    } else {
      wait_asynccnt_0();
    }
    __syncthreads();

    // ---- WMMA over this chunk; each wave owns a 64-wide K slice ----
    const float* tA = &tileA[cnk & 1][0];
    const int kb = cnk * CH;
    const int kk0 = wave * 64;
#pragma unroll
    for (int s = 0; s < 16; ++s) {
      const int k0 = kk0 + s * 4;
      v2f a = *(const v2f*)&tA[aTok * LDW + k0 + kShift];
      const int kg = kb + k0 + kShift;                 // global K of lane's 1st elem
      const float* bp = wpack + ((size_t)(kg >> 1) * 32) * 2;
      v2f b0 = *(const v2f*)(bp + (size_t)bN * 2);
      v2f b1 = *(const v2f*)(bp + (size_t)(16 + bN) * 2);
      c0 = __builtin_amdgcn_wmma_f32_16x16x4_f32(false, a, false, b0,
                                                 (short)0, c0, false, false);
      c1 = __builtin_amdgcn_wmma_f32_16x16x4_f32(false, a, false, b1,
                                                 (short)0, c1, false, false);
      // every element of the 16x4096 tile passes through A exactly once
      // across the 4 waves -> accumulate sum of squares here
      my_ssq += a.x * a.x + a.y * a.y;
    }
    __syncthreads();
  }

  // ---- cross-wave reductions ----
  atomicAdd(&ssq[aTok], my_ssq);
  // C lane layout: vgpr r -> M = r (+8 for upper half-wave), N = lane&15
  const int cm0 = (lane >> 4) * 8;
#pragma unroll
  for (int r = 0; r < 8; ++r) {
    atomicAdd(&cred[(cm0 + r) * 16 + bN],        c0[r]);
    atomicAdd(&cred[256 + (cm0 + r) * 16 + bN],  c1[r]);
  }
  __syncthreads();

  // ---- per-token finalize: lanes 0..15 of wave 0 ----
  if (tid < 16) {
    const int m = tid;
    const float rn = 64.0f / fmaxf(sqrtf(ssq[m]), 1e-12f);   // sqrt(4096)/||h||
    const float a_res = alpha_res[0], a_pre = alpha_pre[0], a_post = alpha_post[0];

    float L[NCH];
#pragma unroll
    for (int c = 0; c < 16; ++c)
      L[c] = beta_res[c] + a_res * rn * cred[m * 16 + c];
#pragma unroll
    for (int c = 0; c < 4; ++c)
      L[16 + c] = beta_pre[c] + a_pre * rn * cred[256 + m * 16 + c];
#pragma unroll
    for (int c = 0; c < 4; ++c)
      L[20 + c] = beta_post[c] + a_post * rn * cred[256 + m * 16 + 4 + c];

    // Sinkhorn in log space, tau = 0.05
    float Z[4][4];
#pragma unroll
    for (int i = 0; i < 4; ++i)
#pragma unroll
      for (int j = 0; j < 4; ++j) Z[i][j] = L[i * 4 + j] * 20.0f;
    float u[4] = {0, 0, 0, 0}, v[4] = {0, 0, 0, 0};
    for (int it = 0; it < 10; ++it) {
#pragma unroll
      for (int i = 0; i < 4; ++i)
        u[i] = -lse4(Z[i][0] + v[0], Z[i][1] + v[1], Z[i][2] + v[2], Z[i][3] + v[3]);
#pragma unroll
      for (int j = 0; j < 4; ++j)
        v[j] = -lse4(Z[0][j] + u[0], Z[1][j] + u[1], Z[2][j] + u[2], Z[3][j] + u[3]);
    }

    float H[NCH];
#pragma unroll
    for (int i = 0; i < 4; ++i)
#pragma unroll
      for (int j = 0; j < 4; ++j) H[i * 4 + j] = expf(Z[i][j] + u[i] + v[j]);

    // H_pre = softmax(L[16..19])
    float mp = fmaxf(fmaxf(L[16], L[17]), fmaxf(L[18], L[19]));
    float e0 = expf(L[16] - mp), e1 = expf(L[17] - mp),
          e2 = expf(L[18] - mp), e3 = expf(L[19] - mp);
    float inv = 1.0f / (e0 + e1 + e2 + e3);
    H[16] = e0 * inv; H[17] = e1 * inv; H[18] = e2 * inv; H[19] = e3 * inv;

    // H_post = 2*sigmoid(L[20..23])
#pragma unroll
    for (int c = 0; c < 4; ++c) H[20 + c] = 2.0f / (1.0f + expf(-L[20 + c]));

    float* dst = hbuf + (tok0 + m) * NCH;
#pragma unroll
    for (int c = 0; c < NCH; ++c) dst[c] = H[c];
  }
}

// ---------------------------------------------------------------------------
// Kernel 2: apply. One block per token, fully coalesced float4 streaming.
// ---------------------------------------------------------------------------
__global__ __launch_bounds__(256) void k_apply(const float* __restrict__ resid,
                                               const float* __restrict__ hbuf,
                                               float* __restrict__ out) {
  __shared__ float H[NCH];
  const size_t tok = blockIdx.x;
  if (threadIdx.x < NCH) H[threadIdx.x] = hbuf[tok * NCH + threadIdx.x];
  __syncthreads();

  const float* rb = resid + tok * SD;
  float* ob = out + tok * SD;
  const int dofs = threadIdx.x * 4;

  float rr[4][4];
#pragma unroll
  for (int s = 0; s < 4; ++s) {
    const float4 t = *(const float4*)(rb + s * D_ + dofs);
    rr[s][0] = t.x; rr[s][1] = t.y; rr[s][2] = t.z; rr[s][3] = t.w;
  }

  float br[4];
#pragma unroll
  for (int q = 0; q < 4; ++q)
    br[q] = H[16] * rr[0][q] + H[17] * rr[1][q] + H[18] * rr[2][q] + H[19] * rr[3][q];

#pragma unroll
  for (int i = 0; i < 4; ++i) {
    float o[4];
#pragma unroll
    for (int q = 0; q < 4; ++q)
      o[q] = H[i * 4 + 0] * rr[0][q] + H[i * 4 + 1] * rr[1][q] +
             H[i * 4 + 2] * rr[2][q] + H[i * 4 + 3] * rr[3][q] +
             H[20 + i] * br[q];
    *(float4*)(ob + i * D_ + dofs) = make_float4(o[0], o[1], o[2], o[3]);
  }
}

// ---------------------------------------------------------------------------
extern "C" void kernel_launch(void* const* d_in, const int* in_sizes, int n_in,
                              void* d_out, int out_size, void* d_ws, size_t ws_size,
                              hipStream_t stream) {
  const float* resid      = (const float*)d_in[0];
  const float* gamma      = (const float*)d_in[1];
  const float* w_res      = (const float*)d_in[2];
  const float* w_pre      = (const float*)d_in[3];
  const float* w_post     = (const float*)d_in[4];
  const float* beta_res   = (const float*)d_in[5];
  const float* beta_pre   = (const float*)d_in[6];
  const float* beta_post  = (const float*)d_in[7];
  const float* alpha_res  = (const float*)d_in[8];
  const float* alpha_pre  = (const float*)d_in[9];
  const float* alpha_post = (const float*)d_in[10];

  float* wpack = (float*)d_ws;                 // 512 KB
  float* hbuf  = wpack + WPACK_FLOATS;         // 8192*24 floats = 768 KB
  float* outp  = (float*)d_out;

  k_fold<<<(KPAIRS * 32 + 255) / 256, 256, 0, stream>>>(gamma, w_res, w_pre, w_post, wpack);
  k_stats<<<NT / 16, 128, 0, stream>>>(resid, wpack, beta_res, beta_pre, beta_post,
                                       alpha_res, alpha_pre, alpha_post, hbuf);
  k_apply<<<NT, 256, 0, stream>>>(resid, hbuf, outp);
}